// QuantumAnnealingOptimizer_89154931130745
// MI455X (gfx1250) — compile-verified
//
#include <hip/hip_runtime.h>
#include <math.h>

typedef __attribute__((ext_vector_type(2))) float v2f;
typedef __attribute__((ext_vector_type(8))) float v8f;

#define ANNEAL_DT 0.01f
#define KC  32           // K-chunk staged in LDS per iteration
#define QP  160          // LDS stride per k-pair row (128 data + 32 pad floats)
                         // 160 mod 64 == 32  ->  ds_load_b64 is bank-conflict free

// ---------------- deterministic counter-based RNG ----------------
__device__ __forceinline__ unsigned hash_u32(unsigned x) {
    x ^= x >> 16; x *= 0x7feb352du;
    x ^= x >> 15; x *= 0x846ca68bu;
    x ^= x >> 16;
    return x;
}
__device__ __forceinline__ float u01(unsigned h) {
    return (float)(h >> 8) * (1.0f / 16777216.0f);
}
__device__ __forceinline__ float gauss1(unsigned step, unsigned idx) {
    unsigned s0 = hash_u32(idx * 0x9E3779B9u + (step + 1u) * 0x85EBCA6Bu + 0x1234567u);
    unsigned s1 = hash_u32(s0 + 0xC2B2AE35u + idx);
    float u1 = u01(s0) + 5.96e-8f;   // avoid log(0)
    float u2 = u01(s1);
    float r = sqrtf(-2.0f * logf(u1));
    return r * __cosf(6.28318530718f * u2);
}

// ---------------- reductions (wave32, deterministic tree) ----------------
__device__ __forceinline__ float wave_reduce_sum(float v) {
#pragma unroll
    for (int off = 16; off > 0; off >>= 1) v += __shfl_down(v, off, 32);
    return v;
}
__device__ __forceinline__ float block_reduce_sum(float v, float* sbuf) {
    const int lane = threadIdx.x & 31;
    const int w    = threadIdx.x >> 5;
    const int nw   = blockDim.x >> 5;
    v = wave_reduce_sum(v);
    __syncthreads();
    if (lane == 0) sbuf[w] = v;
    __syncthreads();
    float r = 0.0f;
    if (w == 0) {
        r = (lane < nw) ? sbuf[lane] : 0.0f;
        r = wave_reduce_sum(r);
        if (lane == 0) sbuf[0] = r;
    }
    __syncthreads();
    float out = sbuf[0];
    __syncthreads();
    return out;
}

// ---------------- Kernel A: P = S @ Q via V_WMMA_F32_16X16X4_F32 ----------------
// block = 256 threads = 8 waves; block computes 128 rows x 64 cols of P.
// Q chunk (KC x 64) is scattered into a k-pair-interleaved LDS layout by
// GLOBAL_LOAD_ASYNC_TO_LDS_B32 (ASYNCcnt), double-buffered so WMMAs on chunk c
// overlap the async fill of chunk c+1. Element (k,n) -> (k>>1)*QP + 2n + (k&1),
// so each B fragment is ONE aligned ds_load_b64 with zero packing moves.
// grid = (D/64, B/128)
__global__ void __launch_bounds__(256)
gemm_sq_wmma(const float* __restrict__ S, const float* __restrict__ Q,
             float* __restrict__ P, int D)
{
    __shared__ __align__(16) float qs[2][(KC / 2) * QP];

    const int tid  = threadIdx.x;
    const int lane = tid & 31;
    const int wave = tid >> 5;
    const int half = lane >> 4;     // 0: K pair {0,1}, 1: K pair {2,3}
    const int l    = lane & 15;
    const int m0   = blockIdx.y * 128 + wave * 16;
    const int n0   = blockIdx.x * 64;

    const float* __restrict__ Srow = S + (size_t)(m0 + l) * (size_t)D;

    v8f acc0 = {}, acc1 = {}, acc2 = {}, acc3 = {};

    // async scatter-fill: 256 threads x 8 B32 packets = KC x 64 floats
    auto issue_chunk = [&](int k0, int buf) {
#pragma unroll
        for (int i = 0; i < 8; ++i) {
            const int e = i * 256 + tid;
            const int k = e >> 6;           // 0..KC-1
            const int n = e & 63;           // 0..63
            unsigned long long g =
                (unsigned long long)(size_t)(Q + (size_t)(k0 + k) * D + n0 + n);
            unsigned lofs =
                (unsigned)(size_t)(&qs[buf][(k >> 1) * QP + 2 * n + (k & 1)]);
            asm volatile("global_load_async_to_lds_b32 %0, %1, off"
                         :: "v"(lofs), "v"(g) : "memory");
        }
    };

    issue_chunk(0, 0);
    asm volatile("s_wait_asynccnt 0x0" ::: "memory");
    __syncthreads();

    const int nchunks = D / KC;
    for (int c = 0; c < nchunks; ++c) {
        const int buf = c & 1;
        if (c + 1 < nchunks) issue_chunk((c + 1) * KC, buf ^ 1);

        const float* __restrict__ qb = &qs[buf][0];
        const int k0 = c * KC;
#pragma unroll
        for (int kk = 0; kk < KC; kk += 4) {
            const int ka = kk + 2 * half;   // even
            v2f a;
            a.x = Srow[k0 + ka];
            a.y = Srow[k0 + ka + 1];
            // (Q[ka][col], Q[ka+1][col]) are adjacent -> one ds_load_b64 each
            const v2f* __restrict__ qrow =
                (const v2f*)(qb + (ka >> 1) * QP + 2 * l);
            v2f b0 = qrow[0];
            v2f b1 = qrow[16];
            v2f b2 = qrow[32];
            v2f b3 = qrow[48];
            acc0 = __builtin_amdgcn_wmma_f32_16x16x4_f32(false, a, false, b0, (short)0, acc0, false, false);
            acc1 = __builtin_amdgcn_wmma_f32_16x16x4_f32(false, a, false, b1, (short)0, acc1, false, false);
            acc2 = __builtin_amdgcn_wmma_f32_16x16x4_f32(false, a, false, b2, (short)0, acc2, false, false);
            acc3 = __builtin_amdgcn_wmma_f32_16x16x4_f32(false, a, false, b3, (short)0, acc3, false, false);
        }

        // chunk c+1 arrived AND every wave is done reading buf before reuse
        asm volatile("s_wait_asynccnt 0x0" ::: "memory");
        __syncthreads();
    }

#pragma unroll
    for (int r = 0; r < 8; ++r) {
        const int row = m0 + r + 8 * half;
        float* __restrict__ prow = P + (size_t)row * (size_t)D + n0 + l;
        prow[0]  = acc0[r];
        prow[16] = acc1[r];
        prow[32] = acc2[r];
        prow[48] = acc3[r];
    }
}

// ---------------- Kernel B: per-row H, ef, noisy update, accept term ----------------
__global__ void __launch_bounds__(256)
row_step(const float* __restrict__ S, const float* __restrict__ P,
         const float* __restrict__ target, const float* __restrict__ lin,
         const float* __restrict__ mixw, const float* __restrict__ namp,
         float* __restrict__ NS, float* __restrict__ edv,
         int D, int step, int num_steps)
{
    __shared__ float sbuf[8];
    const int b = blockIdx.x;
    const size_t base = (size_t)b * (size_t)D;

    float quad = 0.f, li = 0.f, pen = 0.f, mix = 0.f, ssum = 0.f;
    for (int d = threadIdx.x; d < D; d += blockDim.x) {
        const float sv = S[base + d];
        const float pv = P[base + d];
        const float tv = target[base + d];
        quad += sv * pv;
        li   += sv * lin[d];
        const float dd = sv - tv;
        pen  += dd * dd;
        mix  += sv * mixw[d];
        ssum += sv;
    }
    quad = block_reduce_sum(quad, sbuf);
    li   = block_reduce_sum(li,   sbuf);
    pen  = block_reduce_sum(pen,  sbuf);
    mix  = block_reduce_sum(mix,  sbuf);
    ssum = block_reduce_sum(ssum, sbuf);

    const float sfrac = (num_steps > 1) ? (float)step / (float)(num_steps - 1) : 0.0f;
    const float a     = 1.0f - expf(-5.0f * sfrac);
    const float H     = (1.0f - a) * mix + a * (quad + li + pen);
    const float ef    = cosf(H * ANNEAL_DT) - sinf(H * ANNEAL_DT);
    const float nstr  = namp[0] * (1.0f - a);

    float nsum = 0.f;
    for (int d = threadIdx.x; d < D; d += blockDim.x) {
        const float nv = ef * S[base + d] + gauss1((unsigned)step, (unsigned)(base + d)) * nstr;
        NS[base + d] = nv;
        nsum += nv;
    }
    nsum = block_reduce_sum(nsum, sbuf);

    if (threadIdx.x == 0) {
        const float ediff = H * (nsum - ssum);
        const float beta  = 1.0f / (1.0f - a + 1e-8f);
        edv[b] = expf(-beta * fmaxf(ediff, 0.0f));
    }
}

// ---------------- Kernel C: single-block accept decision ----------------
__global__ void __launch_bounds__(256)
decide_k(const float* __restrict__ edv, float* __restrict__ flag, int B, int step)
{
    __shared__ float sbuf[8];
    float v = 0.f;
    for (int i = threadIdx.x; i < B; i += blockDim.x) v += edv[i];
    const float tot = block_reduce_sum(v, sbuf);
    if (threadIdx.x == 0) {
        const float accept = tot / (float)B;
        const float u = u01(hash_u32(0xACCE55u + (unsigned)step * 0x9E3779B9u));
        flag[0] = (u < accept) ? 1.0f : 0.0f;
    }
}

// ---------------- Kernel D: select state, write history, per-row distance ----------------
__global__ void __launch_bounds__(256)
select_k(float* __restrict__ S, const float* __restrict__ NS,
         const float* __restrict__ target, const float* __restrict__ flag,
         float* __restrict__ hist, float* __restrict__ dist, int D)
{
    __shared__ float sbuf[8];
    const int b = blockIdx.x;
    const size_t base = (size_t)b * (size_t)D;
    const float acc = flag[0];

    float pen = 0.f;
    for (int d = threadIdx.x; d < D; d += blockDim.x) {
        const size_t i = base + d;
        const float v = (acc != 0.0f) ? NS[i] : S[i];
        S[i]    = v;
        hist[i] = v;
        const float dd = v - target[i];
        pen += dd * dd;
    }
    pen = block_reduce_sum(pen, sbuf);
    if (threadIdx.x == 0) dist[b] = sqrtf(pen);
}

// ---------------- Kernel E: single-block mean distance -> energies[step] ----------------
__global__ void __launch_bounds__(256)
energy_k(const float* __restrict__ dist, float* __restrict__ energy_out, int B)
{
    __shared__ float sbuf[8];
    float v = 0.f;
    for (int i = threadIdx.x; i < B; i += blockDim.x) v += dist[i];
    const float tot = block_reduce_sum(v, sbuf);
    if (threadIdx.x == 0) energy_out[0] = tot / (float)B;
}

extern "C" void kernel_launch(void* const* d_in, const int* in_sizes, int n_in,
                              void* d_out, int out_size, void* d_ws, size_t ws_size,
                              hipStream_t stream)
{
    const float* init   = (const float*)d_in[0];   // [B,D]
    const float* target = (const float*)d_in[1];   // [B,D]
    const float* Q      = (const float*)d_in[2];   // [D,D]
    const float* lin    = (const float*)d_in[3];   // [D]
    const float* mixw   = (const float*)d_in[4];   // [D]
    const float* namp   = (const float*)d_in[5];   // scalar (device)
    // d_in[6] = num_steps (device scalar) -- recovered from out_size on host.

    const int D  = in_sizes[3];
    const int B  = in_sizes[0] / D;
    const int BD = B * D;
    const int num_steps = (out_size - BD) / (BD + 1);

    float* out       = (float*)d_out;
    float* hist_base = out + BD;                              // [steps, B, D]
    float* energies  = out + BD + (size_t)num_steps * BD;     // [steps]

    float* S    = (float*)d_ws;       // [B,D] evolving state
    float* NS   = S  + BD;            // [B,D] proposed state
    float* P    = NS + BD;            // [B,D] S @ Q
    float* edv  = P  + BD;            // [B]
    float* dist = edv + B;            // [B]
    float* flag = dist + B;           // [1]

    hipMemcpyAsync(S, init, (size_t)BD * sizeof(float),
                   hipMemcpyDeviceToDevice, stream);

    dim3 gemmGrid(D / 64, B / 128);
    for (int step = 0; step < num_steps; ++step) {
        gemm_sq_wmma<<<gemmGrid, 256, 0, stream>>>(S, Q, P, D);
        row_step<<<B, 256, 0, stream>>>(S, P, target, lin, mixw, namp,
                                        NS, edv, D, step, num_steps);
        decide_k<<<1, 256, 0, stream>>>(edv, flag, B, step);
        select_k<<<B, 256, 0, stream>>>(S, NS, target, flag,
                                        hist_base + (size_t)step * BD, dist, D);
        energy_k<<<1, 256, 0, stream>>>(dist, energies + step, B);
    }

    hipMemcpyAsync(out, S, (size_t)BD * sizeof(float),
                   hipMemcpyDeviceToDevice, stream);
}